// EMG2PhonemeAligner_33758442946946
// MI455X (gfx1250) — compile-verified
//
#include <hip/hip_runtime.h>
#include <stdint.h>

typedef __attribute__((ext_vector_type(2))) float v2f;
typedef __attribute__((ext_vector_type(8))) float v8f;

#define Bn 16
#define Tn 8192
#define Cn 512
#define Yn 256
#define YT 16          // phoneme rows per workgroup tile
#define CHUNK 24       // frames per LDS chunk (multiple of 4)
#define PITCH 516      // padded floats per frame row in LDS (bank-conflict pad)
#define BUFELEMS (CHUNK * PITCH)

// Issue async global->LDS DMA for one chunk of frames [kc, kc+CHUNK) into sbuf.
// OOB frames are address-clamped to t1-1: their data is finite and multiplied
// by an exact 0.0 A-weight, so they contribute nothing.
__device__ __forceinline__ void stage_chunk(const float* __restrict__ ebase,
                                            float* sbuf, int tid, int kc, int t1) {
    // CHUNK*Cn/4 = 3072 float4 transfers / 256 threads = 12 each
    #pragma unroll
    for (int it = 0; it < (CHUNK * (Cn / 4)) / 256; ++it) {
        const int idx = tid + (it << 8);
        const int f   = idx >> 7;           // frame within chunk (128 f4/row)
        const int c4  = idx & 127;          // float4 column
        const int t   = kc + f;
        const int tcl = t < t1 ? t : (t1 - 1);
        const float* gp = ebase + (size_t)tcl * Cn + (c4 << 2);
        const uint32_t loff =
            (uint32_t)(uintptr_t)(&sbuf[f * PITCH + (c4 << 2)]);
        asm volatile("global_load_async_to_lds_b128 %0, %1, off"
                     :: "v"(loff), "v"(gp) : "memory");
    }
}

__global__ __launch_bounds__(256)
void pool_by_dur_wmma(const float* __restrict__ emg,
                      const int*  __restrict__ dur,
                      float* __restrict__ out)
{
    const int tid  = threadIdx.x;
    const int lane = tid & 31;
    const int wave = tid >> 5;
    const int b    = blockIdx.x >> 4;           // Y/YT = 16 ytiles per batch
    const int y0   = (blockIdx.x & 15) * YT;

    __shared__ float smem[2 * BUFELEMS];        // double-buffered, ~99 KB
    __shared__ int   ls[YT], le[YT];
    __shared__ float linv[YT];

    // ---- per-row spans: serial prefix over durations (tiny, 16 lanes) ----
    if (tid < YT) {
        const int* drow = dur + b * Yn;
        int cum = 0;
        const int ylast = y0 + tid;
        for (int y = 0; y <= ylast; ++y) cum += drow[y];
        const int d  = drow[ylast];
        const int st = cum - d;
        int sc  = st  < Tn ? st  : Tn;
        int ec  = cum < Tn ? cum : Tn;
        int cnt = ec - sc; if (cnt < 0) cnt = 0;
        ls[tid]   = sc;
        le[tid]   = sc + cnt;
        linv[tid] = cnt > 0 ? 1.0f / (float)cnt : 0.0f;
    }
    __syncthreads();

    const int t0 = ls[0];
    const int t1 = le[YT - 1];

    // per-lane A-matrix row parameters (fixed for whole kernel)
    // A 16x4 layout: lane m = lane&15 is row M; VGPR j holds K = 2*(lane>>4)+j
    const int m  = lane & 15;
    const int kk = (lane >> 4) << 1;            // 0 or 2
    const int sm = ls[m];
    const int em = le[m];

    v8f acc0 = {}, acc1 = {}, acc2 = {}, acc3 = {};

    const float* __restrict__ ebase = emg + (size_t)b * Tn * Cn;

    const int nch = (t1 > t0) ? (t1 - t0 + CHUNK - 1) / CHUNK : 0;
    if (nch > 0) stage_chunk(ebase, smem, tid, t0, t1);

    for (int ci = 0; ci < nch; ++ci) {
        // chunk ci's DMA complete + all waves done reading the buffer we are
        // about to overwrite with the ci+1 prefetch (single barrier covers both)
        asm volatile("s_wait_asynccnt 0" ::: "memory");
        __syncthreads();

        const float* cur = smem + (ci & 1) * BUFELEMS;
        if (ci + 1 < nch)                        // prefetch next chunk (overlaps WMMA)
            stage_chunk(ebase, smem + ((ci + 1) & 1) * BUFELEMS,
                        tid, t0 + (ci + 1) * CHUNK, t1);

        // ---- WMMA accumulate this chunk; wave owns 4 column tiles ----
        const int kc   = t0 + ci * CHUNK;
        const int kend = (t1 - kc) < CHUNK ? (t1 - kc) : CHUNK;
        for (int kl = 0; kl < kend; kl += 4) {
            const int tg = kc + kl + kk;        // global frame of A column kk
            v2f a;
            a.x = (tg     >= sm && tg     < em) ? 1.0f : 0.0f;
            a.y = (tg + 1 >= sm && tg + 1 < em) ? 1.0f : 0.0f;
            const int tl = kl + kk;             // local frame of B row kk
            const float* r0 = &cur[tl * PITCH];
            const float* r1 = &cur[(tl + 1) * PITCH];
            v2f bb; int cb;
            // B 4x16 layout: lane n = lane&15 is column N; VGPR j holds K = kk+j
            cb = ((wave +  0) << 4) + m;
            bb.x = r0[cb]; bb.y = r1[cb];
            acc0 = __builtin_amdgcn_wmma_f32_16x16x4_f32(false, a, false, bb,
                                                         (short)0, acc0, false, false);
            cb = ((wave +  8) << 4) + m;
            bb.x = r0[cb]; bb.y = r1[cb];
            acc1 = __builtin_amdgcn_wmma_f32_16x16x4_f32(false, a, false, bb,
                                                         (short)0, acc1, false, false);
            cb = ((wave + 16) << 4) + m;
            bb.x = r0[cb]; bb.y = r1[cb];
            acc2 = __builtin_amdgcn_wmma_f32_16x16x4_f32(false, a, false, bb,
                                                         (short)0, acc2, false, false);
            cb = ((wave + 24) << 4) + m;
            bb.x = r0[cb]; bb.y = r1[cb];
            acc3 = __builtin_amdgcn_wmma_f32_16x16x4_f32(false, a, false, bb,
                                                         (short)0, acc3, false, false);
        }
    }

    __syncthreads();

    // ---- scale by 1/cnt (sum/cnt like the reference) and store D tiles ----
    // C/D 16x16 layout: lanes 0-15: VGPR r -> (M=r, N=lane); lanes 16-31: M=r+8
    const int mhalf = (lane >> 4) << 3;         // 0 or 8
    float* orow = out + ((size_t)b * Yn + y0) * Cn;
    #pragma unroll
    for (int r = 0; r < 8; ++r) {
        const int M = r + mhalf;
        const float w = linv[M];
        float* op = orow + (size_t)M * Cn + m;
        op[(wave +  0) << 4] = acc0[r] * w;
        op[(wave +  8) << 4] = acc1[r] * w;
        op[(wave + 16) << 4] = acc2[r] * w;
        op[(wave + 24) << 4] = acc3[r] * w;
    }
}

extern "C" void kernel_launch(void* const* d_in, const int* in_sizes, int n_in,
                              void* d_out, int out_size, void* d_ws, size_t ws_size,
                              hipStream_t stream) {
    const float* emg = (const float*)d_in[0];   // (B, T, C) f32
    const int*   dur = (const int*)d_in[1];     // (B, Y) i32
    float* out = (float*)d_out;                 // (B, Y, C) f32
    dim3 grid(Bn * (Yn / YT));                  // 256 workgroups
    pool_by_dur_wmma<<<grid, 256, 0, stream>>>(emg, dur, out);
}